// TwoLayerChebNet_31404800868553
// MI455X (gfx1250) — compile-verified
//
#include <hip/hip_runtime.h>

typedef __attribute__((ext_vector_type(2))) float v2f;
typedef __attribute__((ext_vector_type(8))) float v8f;

#define NODES   100000
#define FDIM    256
#define TILES_M (NODES / 16)   // 6250, exact

// Native CDNA5 fp32 atomic add, non-returning (STOREcnt), device scope so the
// read-modify-write executes in L2's atomic units (visible across WGPs).
__device__ __forceinline__ void atomic_add_f32_dev(float* p, float v) {
    asm volatile("global_atomic_add_f32 %0, %1, off scope:SCOPE_DEV"
                 :: "v"(p), "v"(v)
                 : "memory");
}

// ---------------- de-interleave W (H x 768, feature-major K=3) -> Wk[t] (H x 256) ----------------
__global__ void deinterleave_w(const float* __restrict__ W, float* __restrict__ Wk, int H) {
    int idx = blockIdx.x * blockDim.x + threadIdx.x;
    int total = H * 256 * 3;
    if (idx >= total) return;
    int c = idx % 256;
    int h = (idx / 256) % H;
    int t = idx / (256 * H);
    Wk[((size_t)t * H + h) * 256 + c] = W[(size_t)h * 768 + (size_t)c * 3 + t];
}

// ---------------- B = -A (vectorized), used to fold T2 = 2*L*T1 - T0 ----------------
__global__ void negcopy4(const float4* __restrict__ A, float4* __restrict__ B, int n4) {
    int i = blockIdx.x * blockDim.x + threadIdx.x;
    if (i >= n4) return;
    float4 t = A[i];
    B[i] = make_float4(-t.x, -t.y, -t.z, -t.w);
}

// ---------------- SpMM: y[rows[e],:] += alpha * vals[e] * x[cols[e],:]  (one wave per edge) ----------------
__global__ void spmm_kernel(const int* __restrict__ rows, const int* __restrict__ cols,
                            const float* __restrict__ vals, const float* __restrict__ x,
                            float* __restrict__ y, float alpha, int E) {
    int wave = (blockIdx.x * blockDim.x + threadIdx.x) >> 5;
    int lane = threadIdx.x & 31;
    if (wave >= E) return;
    int r = rows[wave];
    int c = cols[wave];
    float v = alpha * vals[wave];
    const float4* xs = (const float4*)(x + (size_t)c * FDIM);
    float* yr = y + (size_t)r * FDIM;
#pragma unroll
    for (int j = 0; j < 2; ++j) {
        int idx = lane + j * 32;          // 64 float4 = 256 floats per row
        float4 t = xs[idx];
        float* p = yr + idx * 4;
        atomic_add_f32_dev(p + 0, v * t.x);
        atomic_add_f32_dev(p + 1, v * t.y);
        atomic_add_f32_dev(p + 2, v * t.z);
        atomic_add_f32_dev(p + 3, v * t.w);
    }
}

// ---------------- Cheb GEMM: out = act( sum_t A_t @ W_t^T + bias ) via V_WMMA_F32_16X16X4_F32 ----------------
// Each wave: 16 rows x 64 cols (4 v8f accumulators), K loop over 3 terms x 256.
__global__ void cheb_gemm(const float* __restrict__ A0, const float* __restrict__ A1,
                          const float* __restrict__ A2, const float* __restrict__ Wk,
                          const float* __restrict__ bias, float* __restrict__ out,
                          int ncols, int relu_flag) {
    int wave = blockIdx.x * (blockDim.x >> 5) + (threadIdx.x >> 5);
    int lane = threadIdx.x & 31;
    int ngroups = ncols >> 6;                  // groups of 64 output cols
    int totalWaves = TILES_M * ngroups;
    if (wave >= totalWaves) return;            // wave-uniform branch, EXEC stays all-1s

    int tileM = wave % TILES_M;
    int group = wave / TILES_M;
    int half  = lane >> 4;                     // 0: K={0,1} / M rows 0..7 ; 1: K={2,3} / M rows 8..15
    int l16   = lane & 15;
    int rowA  = tileM * 16 + l16;

    const float* Ap[3] = { A0 + (size_t)rowA * FDIM,
                           A1 + (size_t)rowA * FDIM,
                           A2 + (size_t)rowA * FDIM };
    int colbase = group * 64;
    v8f acc0 = {}, acc1 = {}, acc2 = {}, acc3 = {};

    for (int t = 0; t < 3; ++t) {
        const float* Arow = Ap[t];
        // W_t rows for this lane's column within each 16-wide tile
        const float* w0 = Wk + ((size_t)t * ncols + colbase +  0 + l16) * FDIM;
        const float* w1 = Wk + ((size_t)t * ncols + colbase + 16 + l16) * FDIM;
        const float* w2 = Wk + ((size_t)t * ncols + colbase + 32 + l16) * FDIM;
        const float* w3 = Wk + ((size_t)t * ncols + colbase + 48 + l16) * FDIM;
#pragma unroll 4
        for (int kk = 0; kk < FDIM; kk += 4) {
            int ko = kk + 2 * half;
            v2f a  = *(const v2f*)(Arow + ko);   // A[M=l16, K=ko..ko+1]
            v2f b0 = *(const v2f*)(w0 + ko);     // B[K=ko..ko+1, N=l16(+0)]
            v2f b1 = *(const v2f*)(w1 + ko);
            v2f b2 = *(const v2f*)(w2 + ko);
            v2f b3 = *(const v2f*)(w3 + ko);
            acc0 = __builtin_amdgcn_wmma_f32_16x16x4_f32(false, a, false, b0, (short)0, acc0, false, false);
            acc1 = __builtin_amdgcn_wmma_f32_16x16x4_f32(false, a, false, b1, (short)0, acc1, false, false);
            acc2 = __builtin_amdgcn_wmma_f32_16x16x4_f32(false, a, false, b2, (short)0, acc2, false, false);
            acc3 = __builtin_amdgcn_wmma_f32_16x16x4_f32(false, a, false, b3, (short)0, acc3, false, false);
        }
    }

    // Epilogue: D vgpr i -> row i + 8*half, col lane%16 (per 16-col tile)
    v8f accs[4] = { acc0, acc1, acc2, acc3 };
#pragma unroll
    for (int g = 0; g < 4; ++g) {
        int col = colbase + g * 16 + l16;
        float bv = bias[col];
#pragma unroll
        for (int i = 0; i < 8; ++i) {
            int row = tileM * 16 + i + half * 8;
            float v = accs[g][i] + bv;
            if (relu_flag) v = fmaxf(v, 0.0f);
            out[(size_t)row * ncols + col] = v;
        }
    }
}

extern "C" void kernel_launch(void* const* d_in, const int* in_sizes, int n_in,
                              void* d_out, int out_size, void* d_ws, size_t ws_size,
                              hipStream_t stream) {
    const float* x    = (const float*)d_in[0];
    const int*   rows = (const int*)d_in[1];
    const int*   cols = (const int*)d_in[2];
    const float* vals = (const float*)d_in[3];
    const float* W1   = (const float*)d_in[4];
    const float* b1   = (const float*)d_in[5];
    const float* W2   = (const float*)d_in[6];
    const float* b2   = (const float*)d_in[7];
    int E = in_sizes[1];

    size_t NF = (size_t)NODES * FDIM;
    float* A   = (float*)d_ws;           // T1 (layer1) / T1' (layer2)
    float* B   = A + NF;                 // T2 / T2'
    float* C   = B + NF;                 // h
    float* W1k = C + NF;                 // 3 * 256 * 256
    float* W2k = W1k + 3 * 256 * 256;    // 3 *  64 * 256
    float* out = (float*)d_out;

    // weight de-interleave (tiny)
    deinterleave_w<<<(3 * 256 * 256 + 255) / 256, 256, 0, stream>>>(W1, W1k, 256);
    deinterleave_w<<<(3 * 64 * 256 + 255) / 256, 256, 0, stream>>>(W2, W2k, 64);

    int spmmBlocks = (E * 32 + 255) / 256;   // one wave per edge, 8 waves/block
    int n4 = (int)(NF / 4);
    int ncBlocks = (n4 + 255) / 256;

    // ---------------- layer 1 Chebyshev ----------------
    hipMemsetAsync(A, 0, NF * sizeof(float), stream);
    spmm_kernel<<<spmmBlocks, 256, 0, stream>>>(rows, cols, vals, x, A, 1.0f, E);   // T1 = L x
    negcopy4<<<ncBlocks, 256, 0, stream>>>((const float4*)x, (float4*)B, n4);       // T2 = -x ...
    spmm_kernel<<<spmmBlocks, 256, 0, stream>>>(rows, cols, vals, A, B, 2.0f, E);   // ... + 2 L T1
    {
        int waves = TILES_M * (256 / 64);
        cheb_gemm<<<(waves + 7) / 8, 256, 0, stream>>>(x, A, B, W1k, b1, C, 256, 1); // h = relu(...)
    }

    // ---------------- layer 2 Chebyshev ----------------
    hipMemsetAsync(A, 0, NF * sizeof(float), stream);
    spmm_kernel<<<spmmBlocks, 256, 0, stream>>>(rows, cols, vals, C, A, 1.0f, E);   // T1' = L h
    negcopy4<<<ncBlocks, 256, 0, stream>>>((const float4*)C, (float4*)B, n4);       // T2' = -h ...
    spmm_kernel<<<spmmBlocks, 256, 0, stream>>>(rows, cols, vals, A, B, 2.0f, E);   // ... + 2 L T1'
    {
        int waves = TILES_M * (64 / 64);
        cheb_gemm<<<(waves + 7) / 8, 256, 0, stream>>>(C, A, B, W2k, b2, out, 64, 0);
    }
}